// PointNet2SSG_40810779246765
// MI455X (gfx1250) — compile-verified
//
#include <hip/hip_runtime.h>
#include <hip/hip_bf16.h>

// ---------------------------------------------------------------------------
// PointNet++ SSG forward for MI455X (gfx1250, wave32).
// Dense MLP math is routed through V_WMMA_F32_16X16X4_F32 (fp32 WMMA),
// with the weight tile staged (zero-padded) in LDS so the K-loop is
// branch-free and EXEC stays all-ones around every WMMA.
// ---------------------------------------------------------------------------

typedef __attribute__((ext_vector_type(2))) float v2f;
typedef __attribute__((ext_vector_type(8))) float v8f;

#define FPS_NT 1024
#define WMAX   8192   // max O*Cpad over all layers (128*64) -> 32KB LDS

// ---------------- Farthest point sampling -----------------------------------
// One workgroup per batch. Point coords + running min-distances live in
// registers (PT points per thread); per-step argmax via LDS tree reduction;
// the selected point's coords are broadcast through LDS by the owning thread
// (no global memory on the serial critical path).
template<int PT>
__global__ __launch_bounds__(FPS_NT)
void fps_kernel(const float* __restrict__ xyz, int n, int npoint,
                int* __restrict__ idxOut)
{
  const int b   = blockIdx.x;
  const int tid = threadIdx.x;
  __shared__ float sd[FPS_NT];
  __shared__ int   si[FPS_NT];
  __shared__ float sLast[3];
  const float* Pb = xyz + (size_t)b * n * 3;

  float px[PT], py[PT], pz[PT], dd[PT];
#pragma unroll
  for (int j = 0; j < PT; ++j) {
    int i = j * FPS_NT + tid;
    if (i < n) { px[j] = Pb[i*3+0]; py[j] = Pb[i*3+1]; pz[j] = Pb[i*3+2]; dd[j] = 1e10f; }
    else       { px[j] = 0.f; py[j] = 0.f; pz[j] = 0.f; dd[j] = -1e30f; }
  }
  if (tid == 0) {
    idxOut[(size_t)b * npoint] = 0;
    sLast[0] = Pb[0]; sLast[1] = Pb[1]; sLast[2] = Pb[2];
  }
  __syncthreads();

  for (int t = 1; t < npoint; ++t) {
    const float lx = sLast[0], ly = sLast[1], lz = sLast[2];
    float best = -1e30f; int bi = 0x7fffffff;
#pragma unroll
    for (int j = 0; j < PT; ++j) {
      int i = j * FPS_NT + tid;
      float dx = px[j]-lx, dy = py[j]-ly, dz = pz[j]-lz;
      float d  = dx*dx + dy*dy + dz*dz;
      if (i < n && d < dd[j]) dd[j] = d;
      float dj = dd[j];
      if (dj > best || (dj == best && i < bi)) { best = dj; bi = i; }
    }
    sd[tid] = best; si[tid] = bi;
    __syncthreads();
    for (int s = FPS_NT/2; s > 0; s >>= 1) {
      if (tid < s) {
        float od = sd[tid+s]; int oi = si[tid+s];
        if (od > sd[tid] || (od == sd[tid] && oi < si[tid])) { sd[tid] = od; si[tid] = oi; }
      }
      __syncthreads();
    }
    const int nxt = si[0];
    if (tid == 0) idxOut[(size_t)b*npoint + t] = nxt;
    if (tid == (nxt & (FPS_NT-1))) {
      const int jj = nxt >> 10;   // nxt / FPS_NT
#pragma unroll
      for (int j = 0; j < PT; ++j)
        if (j == jj) { sLast[0] = px[j]; sLast[1] = py[j]; sLast[2] = pz[j]; }
    }
    __syncthreads();
  }
}

// ---------------- Gather sampled centers ------------------------------------
__global__ void gather_xyz_kernel(const float* __restrict__ xyz,
                                  const int* __restrict__ idx,
                                  int n, int S, int B, float* __restrict__ nxyz)
{
  int t = blockIdx.x * blockDim.x + threadIdx.x;
  if (t >= B * S) return;
  int b = t / S;
  int g = idx[t];
  const float* p = xyz + ((size_t)b * n + g) * 3;
  nxyz[(size_t)t*3+0] = p[0];
  nxyz[(size_t)t*3+1] = p[1];
  nxyz[(size_t)t*3+2] = p[2];
}

// ---------------- Ball query: first K in-radius indices (ascending) ---------
__global__ void ball_query_kernel(const float* __restrict__ xyz,
                                  const float* __restrict__ nxyz,
                                  int n, int S, int K, int B, float r2,
                                  int* __restrict__ gidx)
{
  int t = blockIdx.x * blockDim.x + threadIdx.x;
  if (t >= B * S) return;
  int b = t / S;
  const float* Pb = xyz + (size_t)b * n * 3;
  float cx = nxyz[(size_t)t*3+0], cy = nxyz[(size_t)t*3+1], cz = nxyz[(size_t)t*3+2];
  int* out = gidx + (size_t)t * K;
  int cnt = 0, first = 0;
  for (int j = 0; j < n && cnt < K; ++j) {
    float dx = Pb[j*3+0]-cx, dy = Pb[j*3+1]-cy, dz = Pb[j*3+2]-cz;
    float d = dx*dx + dy*dy + dz*dz;
    if (d < r2) { if (cnt == 0) first = j; out[cnt++] = j; }
  }
  for (int k2 = cnt; k2 < K; ++k2) out[k2] = first;
}

// ---------------- Build grouped SA feature matrix X[Cpad][Ptot] -------------
__global__ void sa_build_kernel(const float* __restrict__ xyz,
                                const float* __restrict__ nxyz,
                                const int* __restrict__ gidx,
                                const float* __restrict__ feat,
                                int n, int S, int K, int Cf, int featPointMajor,
                                float* __restrict__ X, int Cpad, int Ptot)
{
  int col = blockIdx.x * blockDim.x + threadIdx.x;
  if (col >= Ptot) return;
  int sc = col / K;          // b*S + s
  int b  = sc / S;
  int g  = gidx[col];
  const float* Pb = xyz + (size_t)b * n * 3;
  float cx = nxyz[(size_t)sc*3+0], cy = nxyz[(size_t)sc*3+1], cz = nxyz[(size_t)sc*3+2];
  X[(size_t)0*Ptot + col] = Pb[g*3+0] - cx;
  X[(size_t)1*Ptot + col] = Pb[g*3+1] - cy;
  X[(size_t)2*Ptot + col] = Pb[g*3+2] - cz;
  if (featPointMajor) {                       // stage 1: feats == xyz, [n][3]
    const float* F = feat + (size_t)b * n * 3;
    for (int c = 0; c < Cf; ++c) X[(size_t)(3+c)*Ptot + col] = F[g*3+c];
  } else {                                    // [Cf][n] per batch
    const float* F = feat + (size_t)b * Cf * n;
    for (int c = 0; c < Cf; ++c) X[(size_t)(3+c)*Ptot + col] = F[(size_t)c*n + g];
  }
  for (int c = 3 + Cf; c < Cpad; ++c) X[(size_t)c*Ptot + col] = 0.f;  // zero pad rows
}

// ---------------- WMMA fp32 GEMM + scale/bias/ReLU epilogue -----------------
// Y = relu(g * (W[O][C] x X[Cpad][P]) + b). Weights are staged in LDS,
// zero-padded to Cpad, so the K-loop is completely branch-free (EXEC stays
// all-ones around every WMMA). Each wave owns TWO 16x16 output tiles that
// share the A fragment -> 2 back-to-back independent V_WMMA_F32_16X16X4_F32
// per K-step. Block = 4 waves = 8 N-tiles; grid.x = P/128, grid.y = O/16.
// outMode==1 scatters into the (B,96,Mout) concatenated output tensor.
__global__ __launch_bounds__(128)
void wmma_mlp_kernel(const float* __restrict__ W, const float* __restrict__ gain,
                     const float* __restrict__ bias, const float* __restrict__ X,
                     float* __restrict__ Y, int O, int C, int Cpad, int P,
                     int outMode, int chanOff, int Mout)
{
  __shared__ float sW[WMAX];
  __shared__ float sG[128];
  __shared__ float sB[128];

  // Cooperative, padded weight stage (guards live here, outside the hot loop).
  for (int i = threadIdx.x; i < O * Cpad; i += 128) {
    int o = i / Cpad, c = i - o * Cpad;
    sW[i] = (c < C) ? W[(size_t)o * C + c] : 0.f;
  }
  for (int i = threadIdx.x; i < O; i += 128) { sG[i] = gain[i]; sB[i] = bias[i]; }
  __syncthreads();

  const int lane  = threadIdx.x & 31;
  const int wav   = threadIdx.x >> 5;
  const int oBase = blockIdx.y * 16;
  const int row   = lane & 15;               // M index (A) / N index (B)
  const int kh    = (lane >> 4) << 1;        // K half: lanes 0-15 -> K{0,1}, 16-31 -> K{2,3}
  const int p0    = (blockIdx.x * 8 + wav * 2) * 16 + row;  // tile 0 column
  const int p1    = p0 + 16;                                // tile 1 column

  const float* __restrict__ sWrow = sW + (size_t)(oBase + row) * Cpad + kh;
  const float* __restrict__ x0    = X + p0;
  const float* __restrict__ x1    = X + p1;

  v8f acc0 = {};
  v8f acc1 = {};
  for (int c0 = 0; c0 < Cpad; c0 += 4) {
    const size_t rA = (size_t)(c0 + kh) * P;
    const size_t rB = rA + P;
    v2f av  = { sWrow[c0],  sWrow[c0 + 1] };   // ds_load, branch-free
    v2f bv0 = { x0[rA],     x0[rB]       };
    v2f bv1 = { x1[rA],     x1[rB]       };
    acc0 = __builtin_amdgcn_wmma_f32_16x16x4_f32(false, av, false, bv0,
                                                 (short)0, acc0, false, false);
    acc1 = __builtin_amdgcn_wmma_f32_16x16x4_f32(false, av, false, bv1,
                                                 (short)0, acc1, false, false);
  }

  const int half = (lane >> 4) * 8;          // C/D layout: M = v + 8*(lane>=16)
#pragma unroll
  for (int v = 0; v < 8; ++v) {
    const int o = oBase + half + v;
    const float gg = sG[o], bb = sB[o];
    float v0 = fmaxf(acc0[v] * gg + bb, 0.f);
    float v1 = fmaxf(acc1[v] * gg + bb, 0.f);
    if (outMode == 0) {
      Y[(size_t)o * P + p0] = v0;
      Y[(size_t)o * P + p1] = v1;
    } else {
      int b0 = p0 / Mout, m0 = p0 - b0 * Mout;
      int b1 = p1 / Mout, m1 = p1 - b1 * Mout;
      Y[((size_t)b0 * 96 + chanOff + o) * Mout + m0] = v0;
      Y[((size_t)b1 * 96 + chanOff + o) * Mout + m1] = v1;
    }
  }
}

// ---------------- Max over nsample ------------------------------------------
__global__ void maxpool_kernel(const float* __restrict__ Y, float* __restrict__ feats,
                               int O, int S, int K, int B, int Ptot)
{
  int t = blockIdx.x * blockDim.x + threadIdx.x;
  if (t >= B * O * S) return;
  int s = t % S; int r = t / S; int o = r % O; int b = r / O;
  const float* base = Y + (size_t)o * Ptot + ((size_t)b * S + s) * K;
  float m = base[0];
  for (int k = 1; k < K; ++k) m = fmaxf(m, base[k]);
  feats[t] = m;   // t == ((b*O)+o)*S + s
}

// ---------------- 3-NN + inverse-distance weights ---------------------------
__global__ void fp_knn_kernel(const float* __restrict__ qp, const float* __restrict__ kxyz,
                              int S, int M, int B, int* __restrict__ ki, float* __restrict__ kw)
{
  int t = blockIdx.x * blockDim.x + threadIdx.x;
  if (t >= B * M) return;
  int b = t / M; int m = t - b * M;
  float ux = qp[((size_t)b*3+0)*M + m];
  float uy = qp[((size_t)b*3+1)*M + m];
  float uz = qp[((size_t)b*3+2)*M + m];
  const float* Kb = kxyz + (size_t)b * S * 3;
  float d0 = 3.4e38f, d1 = 3.4e38f, d2 = 3.4e38f;
  int   j0 = 0, j1 = 0, j2 = 0;
  for (int j = 0; j < S; ++j) {
    float dx = Kb[j*3+0]-ux, dy = Kb[j*3+1]-uy, dz = Kb[j*3+2]-uz;
    float d = dx*dx + dy*dy + dz*dz;
    if (d < d0)      { d2=d1; j2=j1; d1=d0; j1=j0; d0=d; j0=j; }
    else if (d < d1) { d2=d1; j2=j1; d1=d; j1=j; }
    else if (d < d2) { d2=d; j2=j; }
  }
  float w0 = 1.f/(d0+1e-8f), w1 = 1.f/(d1+1e-8f), w2 = 1.f/(d2+1e-8f);
  float ws = w0 + w1 + w2;
  ki[(size_t)t*3+0] = j0; ki[(size_t)t*3+1] = j1; ki[(size_t)t*3+2] = j2;
  kw[(size_t)t*3+0] = w0/ws; kw[(size_t)t*3+1] = w1/ws; kw[(size_t)t*3+2] = w2/ws;
}

// ---------------- Build FP input: [interp(Cf) ; query(3) ; pad] -------------
__global__ void fp_build_kernel(const float* __restrict__ qp, const float* __restrict__ feats,
                                const int* __restrict__ ki, const float* __restrict__ kw,
                                int Cf, int S, int M, int B,
                                float* __restrict__ X, int Cpad, int P)
{
  int t = blockIdx.x * blockDim.x + threadIdx.x;
  if (t >= B * M) return;
  int b = t / M; int m = t - b * M;
  int i0 = ki[(size_t)t*3+0], i1 = ki[(size_t)t*3+1], i2 = ki[(size_t)t*3+2];
  float w0 = kw[(size_t)t*3+0], w1 = kw[(size_t)t*3+1], w2 = kw[(size_t)t*3+2];
  const float* F = feats + (size_t)b * Cf * S;
  for (int c = 0; c < Cf; ++c) {
    const float* Fc = F + (size_t)c * S;
    X[(size_t)c * P + t] = Fc[i0]*w0 + Fc[i1]*w1 + Fc[i2]*w2;
  }
  for (int d = 0; d < 3; ++d)
    X[(size_t)(Cf + d) * P + t] = qp[((size_t)b*3 + d)*M + m];
  for (int c = Cf + 3; c < Cpad; ++c)
    X[(size_t)c * P + t] = 0.f;
}

// ---------------------------------------------------------------------------
extern "C" void kernel_launch(void* const* d_in, const int* in_sizes, int n_in,
                              void* d_out, int out_size, void* d_ws, size_t ws_size,
                              hipStream_t stream)
{
  (void)in_sizes; (void)n_in; (void)out_size; (void)ws_size;
  const int B = 4, N = 16384, M = 16384;
  const float* surf = (const float*)d_in[0];   // (B,N,3)
  const float* qp   = (const float*)d_in[1];   // (B,3,M)

  const float *saW[3][2], *saG[3][2], *saB[3][2];
  const float *fpW[3][2], *fpG[3][2], *fpB[3][2];
  int a = 2;
  for (int s = 0; s < 3; ++s)
    for (int l = 0; l < 2; ++l) {
      saW[s][l] = (const float*)d_in[a++];
      saG[s][l] = (const float*)d_in[a++];
      saB[s][l] = (const float*)d_in[a++];
    }
  for (int s = 0; s < 3; ++s)
    for (int l = 0; l < 2; ++l) {
      fpW[s][l] = (const float*)d_in[a++];
      fpG[s][l] = (const float*)d_in[a++];
      fpB[s][l] = (const float*)d_in[a++];
    }

  // ---- workspace bump allocation ----
  char* ws = (char*)d_ws;
  size_t off = 0;
  auto bump = [&](size_t bytes) -> char* {
    char* p = ws + off;
    off = (off + bytes + 255) & ~(size_t)255;
    return p;
  };
  int*   idx1 = (int*)  bump((size_t)B*1024   * sizeof(int));
  float* nx1  = (float*)bump((size_t)B*1024*3 * sizeof(float));
  float* f1   = (float*)bump((size_t)B*32*1024* sizeof(float));
  int*   idx2 = (int*)  bump((size_t)B*512    * sizeof(int));
  float* nx2  = (float*)bump((size_t)B*512*3  * sizeof(float));
  float* f2   = (float*)bump((size_t)B*64*512 * sizeof(float));
  int*   idx3 = (int*)  bump((size_t)B*128    * sizeof(int));
  float* nx3  = (float*)bump((size_t)B*128*3  * sizeof(float));
  float* f3   = (float*)bump((size_t)B*128*128* sizeof(float));
  int*   gidx = (int*)  bump((size_t)65536    * sizeof(int));     // max B*S*K
  int*   ki   = (int*)  bump((size_t)B*M*3    * sizeof(int));
  float* kw   = (float*)bump((size_t)B*M*3    * sizeof(float));
  float* Xb   = (float*)bump((size_t)132*65536* sizeof(float));   // max Cpad*P
  float* Hb   = (float*)bump((size_t)64*65536 * sizeof(float));   // layer-1 out
  float* Ob   = (float*)bump((size_t)64*65536 * sizeof(float));   // layer-2 out (pre-max)
  float* outp = (float*)d_out;                                    // (B,96,M)

  const int T = 256;
  auto g1 = [](int n_) { return (n_ + 255) / 256; };

  // =============== SA stage 1: n=16384 -> S=1024, K=16, MLP 6->16->32 =======
  fps_kernel<16><<<B, FPS_NT, 0, stream>>>(surf, N, 1024, idx1);
  gather_xyz_kernel<<<g1(B*1024), T, 0, stream>>>(surf, idx1, N, 1024, B, nx1);
  ball_query_kernel<<<g1(B*1024), T, 0, stream>>>(surf, nx1, N, 1024, 16, B, 0.01f, gidx);
  sa_build_kernel<<<g1(65536), T, 0, stream>>>(surf, nx1, gidx, surf, N, 1024, 16, 3, 1, Xb, 8, 65536);
  wmma_mlp_kernel<<<dim3(65536/128, 1), 128, 0, stream>>>(saW[0][0], saG[0][0], saB[0][0], Xb, Hb, 16, 6, 8, 65536, 0, 0, 0);
  wmma_mlp_kernel<<<dim3(65536/128, 2), 128, 0, stream>>>(saW[0][1], saG[0][1], saB[0][1], Hb, Ob, 32, 16, 16, 65536, 0, 0, 0);
  maxpool_kernel<<<g1(B*32*1024), T, 0, stream>>>(Ob, f1, 32, 1024, 16, B, 65536);

  // =============== SA stage 2: 1024 -> 512, K=32, MLP 35->32->64 ============
  fps_kernel<1><<<B, FPS_NT, 0, stream>>>(nx1, 1024, 512, idx2);
  gather_xyz_kernel<<<g1(B*512), T, 0, stream>>>(nx1, idx2, 1024, 512, B, nx2);
  ball_query_kernel<<<g1(B*512), T, 0, stream>>>(nx1, nx2, 1024, 512, 32, B, 0.04f, gidx);
  sa_build_kernel<<<g1(65536), T, 0, stream>>>(nx1, nx2, gidx, f1, 1024, 512, 32, 32, 0, Xb, 36, 65536);
  wmma_mlp_kernel<<<dim3(65536/128, 2), 128, 0, stream>>>(saW[1][0], saG[1][0], saB[1][0], Xb, Hb, 32, 35, 36, 65536, 0, 0, 0);
  wmma_mlp_kernel<<<dim3(65536/128, 4), 128, 0, stream>>>(saW[1][1], saG[1][1], saB[1][1], Hb, Ob, 64, 32, 32, 65536, 0, 0, 0);
  maxpool_kernel<<<g1(B*64*512), T, 0, stream>>>(Ob, f2, 64, 512, 32, B, 65536);

  // =============== SA stage 3: 512 -> 128, K=64, MLP 67->64->128 ============
  fps_kernel<1><<<B, FPS_NT, 0, stream>>>(nx2, 512, 128, idx3);
  gather_xyz_kernel<<<g1(B*128), T, 0, stream>>>(nx2, idx3, 512, 128, B, nx3);
  ball_query_kernel<<<g1(B*128), T, 0, stream>>>(nx2, nx3, 512, 128, 64, B, 0.16f, gidx);
  sa_build_kernel<<<g1(32768), T, 0, stream>>>(nx2, nx3, gidx, f2, 512, 128, 64, 64, 0, Xb, 68, 32768);
  wmma_mlp_kernel<<<dim3(32768/128, 4), 128, 0, stream>>>(saW[2][0], saG[2][0], saB[2][0], Xb, Hb, 64, 67, 68, 32768, 0, 0, 0);
  wmma_mlp_kernel<<<dim3(32768/128, 8), 128, 0, stream>>>(saW[2][1], saG[2][1], saB[2][1], Hb, Ob, 128, 64, 64, 32768, 0, 0, 0);
  maxpool_kernel<<<g1(B*128*128), T, 0, stream>>>(Ob, f3, 128, 128, 64, B, 32768);

  // =============== FP stage 1: interp from 1024, MLP 35->32->32 -> ch 0 =====
  fp_knn_kernel<<<g1(B*M), T, 0, stream>>>(qp, nx1, 1024, M, B, ki, kw);
  fp_build_kernel<<<g1(B*M), T, 0, stream>>>(qp, f1, ki, kw, 32, 1024, M, B, Xb, 36, 65536);
  wmma_mlp_kernel<<<dim3(65536/128, 2), 128, 0, stream>>>(fpW[0][0], fpG[0][0], fpB[0][0], Xb, Hb, 32, 35, 36, 65536, 0, 0, 0);
  wmma_mlp_kernel<<<dim3(65536/128, 2), 128, 0, stream>>>(fpW[0][1], fpG[0][1], fpB[0][1], Hb, outp, 32, 32, 32, 65536, 1, 0, M);

  // =============== FP stage 2: interp from 512, MLP 67->32->32 -> ch 32 =====
  fp_knn_kernel<<<g1(B*M), T, 0, stream>>>(qp, nx2, 512, M, B, ki, kw);
  fp_build_kernel<<<g1(B*M), T, 0, stream>>>(qp, f2, ki, kw, 64, 512, M, B, Xb, 68, 65536);
  wmma_mlp_kernel<<<dim3(65536/128, 2), 128, 0, stream>>>(fpW[1][0], fpG[1][0], fpB[1][0], Xb, Hb, 32, 67, 68, 65536, 0, 0, 0);
  wmma_mlp_kernel<<<dim3(65536/128, 2), 128, 0, stream>>>(fpW[1][1], fpG[1][1], fpB[1][1], Hb, outp, 32, 32, 32, 65536, 1, 32, M);

  // =============== FP stage 3: interp from 128, MLP 131->32->32 -> ch 64 ====
  fp_knn_kernel<<<g1(B*M), T, 0, stream>>>(qp, nx3, 128, M, B, ki, kw);
  fp_build_kernel<<<g1(B*M), T, 0, stream>>>(qp, f3, ki, kw, 128, 128, M, B, Xb, 132, 65536);
  wmma_mlp_kernel<<<dim3(65536/128, 2), 128, 0, stream>>>(fpW[2][0], fpG[2][0], fpB[2][0], Xb, Hb, 32, 131, 132, 65536, 0, 0, 0);
  wmma_mlp_kernel<<<dim3(65536/128, 2), 128, 0, stream>>>(fpW[2][1], fpG[2][1], fpB[2][1], Hb, outp, 32, 32, 32, 65536, 1, 64, M);
}